// GRUModel_46660524703783
// MI455X (gfx1250) — compile-verified
//
#include <hip/hip_runtime.h>
#include <hip/hip_bf16.h>
#include <math.h>

typedef __attribute__((ext_vector_type(2))) float v2f;
typedef __attribute__((ext_vector_type(8))) float v8f;
typedef unsigned int u32x4 __attribute__((ext_vector_type(4)));
typedef int i32x8 __attribute__((ext_vector_type(8)));
typedef int i32x4 __attribute__((ext_vector_type(4)));

#define Bv 64
#define Tv 512
#define Dv 256
#define Hv 256
#define NHv 768   // 3*H
#define Cv 10
#define SCAN_WG 12
#define SCAN_THREADS 256
#define HP 260    // padded LDS row stride for h (bank-conflict-free b64 reads)

__device__ __forceinline__ v8f wmma_f32(v2f a, v2f b, v8f c) {
  // V_WMMA_F32_16X16X4_F32 : D = A(16x4) * B(4x16) + C(16x16), fp32
  return __builtin_amdgcn_wmma_f32_16x16x4_f32(false, a, false, b, (short)0, c,
                                               false, false);
}

// -------------------------------------------------- TDM: 2-D tile -> LDS
// D# packing per CDNA5 ISA ch.8 (group0: count/lds_addr/global_addr/type=2,
// group1: data_size=4B, tensor/tile dims, dim0 stride, optional LDS padding).
// One issue per WG (TDM ignores EXEC); completion tracked with TENSORcnt.
__device__ __forceinline__ void tdm_load_2d(
    unsigned lds_off, const void* gptr,
    unsigned tensor_d0, unsigned tensor_d1,   // elements / rows
    unsigned tile_d0, unsigned tile_d1,       // tile cols / rows
    unsigned stride0,                         // row pitch (elements)
    unsigned pad_bits) {                      // pad_enable/interval/amount
  const unsigned long long ga = (unsigned long long)(uintptr_t)gptr;
  u32x4 g0;
  g0[0] = 1u;                                            // count=1, load
  g0[1] = lds_off;                                       // LDS byte address
  g0[2] = (unsigned)ga;                                  // global addr lo
  g0[3] = (unsigned)((ga >> 32) & 0x1FFFFFFu) | (2u << 30);  // hi + type=2
  i32x8 g1;
  g1[0] = (int)(0x00020000u | pad_bits);                 // data_size=4B
  g1[1] = (int)((tensor_d0 & 0xFFFFu) << 16);            // dim0 lo16
  g1[2] = (int)((tensor_d0 >> 16) | ((tensor_d1 & 0xFFFFu) << 16));
  g1[3] = (int)((tensor_d1 >> 16) | (tile_d0 << 16));    // tile_dim0
  g1[4] = (int)tile_d1;                                  // tile_dim1 (dim2=0)
  g1[5] = (int)stride0;                                  // dim0 stride lo32
  g1[6] = 0;
  g1[7] = 0;
  const i32x4 z4 = {0, 0, 0, 0};                         // 2-D: groups 2/3 off
  const i32x8 z8 = {0, 0, 0, 0, 0, 0, 0, 0};
  __builtin_amdgcn_tensor_load_to_lds(g0, g1, z4, z4, z8, 0);
}

// ---------------------------------------------------------------- embedding
__global__ void embed_kernel(const int* __restrict__ tokens,
                             const float* __restrict__ emb,
                             float* __restrict__ x) {
  const int bt  = blockIdx.x;              // 0 .. B*T-1
  const int tok = tokens[bt];
  const float4* src = (const float4*)(emb + (size_t)tok * Dv);
  float4*       dst = (float4*)(x + (size_t)bt * Dv);
  dst[threadIdx.x] = src[threadIdx.x];     // 64 threads x 16B = 256 floats
}

// ------------------------------------------------- xproj = x @ W[l] + b_i[l]
// M = B*T = 32768, K = 256, N = 768.  One 16-col N-tile x 64-row M-block per
// wave; B fragment reused across 4 WMMA accumulators.
__global__ __launch_bounds__(256) void xproj_gemm_kernel(
    const float* __restrict__ A,      // [M, 256]
    const float* __restrict__ W,      // [256, 768]
    const float* __restrict__ bias,   // [768]
    float* __restrict__ out) {        // [M, 768]
  const int wave    = threadIdx.x >> 5;
  const int lane    = threadIdx.x & 31;
  const int gw      = blockIdx.x * 8 + wave;
  const int n_tile  = gw % 48;
  const int m_block = gw / 48;
  const int m       = lane & 15;
  const int kb      = (lane >> 4) << 1;           // 0 or 2
  const int nc      = n_tile * 16 + m;            // this lane's B/D column
  const int m0      = m_block * 64;

  v8f acc[4] = {{0.f}, {0.f}, {0.f}, {0.f}};
  for (int k0 = 0; k0 < 256; k0 += 4) {
    const int k = k0 + kb;
    v2f bf;
    bf.x = W[(size_t)k * NHv + nc];
    bf.y = W[(size_t)(k + 1) * NHv + nc];
#pragma unroll
    for (int i = 0; i < 4; ++i) {
      const float2 av = *(const float2*)(A + (size_t)(m0 + i * 16 + m) * 256 + k);
      v2f af; af.x = av.x; af.y = av.y;
      acc[i] = wmma_f32(af, bf, acc[i]);
    }
  }
  const float bv = bias[nc];
#pragma unroll
  for (int i = 0; i < 4; ++i) {
#pragma unroll
    for (int e = 0; e < 8; ++e) {
      const int row = m0 + i * 16 + e + ((lane >> 4) << 3);
      out[(size_t)row * NHv + nc] = acc[i][e] + bv;
    }
  }
}

// ---------------------------------------------------------- grid-wide barrier
__device__ __forceinline__ void grid_barrier(unsigned* bar, unsigned target) {
  __threadfence();                    // publish this thread's writes
  __syncthreads();                    // whole WG done
  if (threadIdx.x == 0) {
    atomicAdd(bar, 1u);
    while (atomicAdd(bar, 0u) < target) { __builtin_amdgcn_s_sleep(1); }
  }
  __syncthreads();
}

__global__ void bar_init_kernel(unsigned* bar) { *bar = 0u; }

// ------------------------------------------------------------- GRU time scan
// Persistent kernel: 12 WGs x 8 waves cover the 64x768 rec matrix (192 tiles,
// 2 per wave).  Per-WG R slice (64KB) DMA'd to LDS once via the Tensor Data
// Mover; h (64KB) DMA'd per step with hardware row-padding to a 260-float
// stride.  Each step:
//   TDM h -> s_wait_tensorcnt -> rec = h @ R + b_r (WMMA from LDS) ->
//   barrier -> gates/state update (VALU) -> barrier.
__global__ __launch_bounds__(256) void gru_scan_kernel(
    const float* __restrict__ xproj,   // [B*T, 768]
    const float* __restrict__ R,       // [256, 768]
    const float* __restrict__ br,      // [768]
    const int*   __restrict__ tokens,  // [B, T]
    float* __restrict__ h,             // [64, 256]
    float* __restrict__ rec,           // [64, 768]
    float* __restrict__ xout,          // [B*T, 256]
    unsigned* __restrict__ bar) {
  __shared__ float ldsR[Hv * 64];      // [k][64 cols of this WG]   64 KB
  __shared__ float ldsH[Bv * HP];      // [row][k], padded stride   65 KB

  const int tid  = threadIdx.x;
  const int wg   = blockIdx.x;
  const int gtid = wg * SCAN_THREADS + tid;
  const int wave = tid >> 5;
  const int lane = tid & 31;
  const int m    = lane & 15;
  const int kb   = (lane >> 4) << 1;
  const int m0   = (wave & 3) * 16;                 // batch-row tile (B=64)
  const int n0a  = wg * 64 + (wave >> 2) * 32;      // first of two col tiles
  const int ca   = n0a + m;
  const int cb   = n0a + 16 + m;
  const int cla  = ca - wg * 64;                    // LDS-local columns
  const int clb  = cla + 16;
  const float bra = br[ca];
  const float brb = br[cb];
  const unsigned ldsR_off = (unsigned)(uintptr_t)&ldsR[0];
  const unsigned ldsH_off = (unsigned)(uintptr_t)&ldsH[0];
  unsigned target = 0;

  // TDM: stage this WG's R slice (256 rows x 64 cols, pitch 768) into LDS.
  if (tid == 0) {
    tdm_load_2d(ldsR_off, R + wg * 64, NHv, Hv, 64, Hv, NHv, 0u);
  }

  // h0 = 0 (each element owned by exactly one thread in the grid)
  for (int idx = gtid; idx < Bv * Hv; idx += SCAN_WG * SCAN_THREADS)
    h[idx] = 0.0f;
  if (tid == 0) __builtin_amdgcn_s_wait_tensorcnt(0);
  target += SCAN_WG; grid_barrier(bar, target);

  for (int t = 0; t < Tv; ++t) {
    // ---- TDM: stage h (64 rows x 256 cols) into LDS with 4-DWORD row pad
    // pad_enable(bit20) | pad_interval=7 (256 DWORDs) | pad_amount=3 (4 DW)
    if (tid == 0) {
      tdm_load_2d(ldsH_off, h, Hv, Bv, Hv, Bv, Hv,
                  (1u << 20) | (7u << 22) | (3u << 25));
      __builtin_amdgcn_s_wait_tensorcnt(0);
    }
    __syncthreads();

    // ---- phase 1: rec = h @ R + b_r  (two 16x16 tiles per wave, K=256)
    v8f acc0 = {0.f};
    v8f acc1 = {0.f};
    for (int k0 = 0; k0 < Hv; k0 += 4) {
      const int k = k0 + kb;
      const float2 av = *(const float2*)(&ldsH[(m0 + m) * HP + k]);
      v2f a; a.x = av.x; a.y = av.y;
      v2f b0, b1;
      b0.x = ldsR[k * 64 + cla];
      b0.y = ldsR[(k + 1) * 64 + cla];
      b1.x = ldsR[k * 64 + clb];
      b1.y = ldsR[(k + 1) * 64 + clb];
      acc0 = wmma_f32(a, b0, acc0);
      acc1 = wmma_f32(a, b1, acc1);
    }
#pragma unroll
    for (int e = 0; e < 8; ++e) {
      const int row = m0 + e + ((lane >> 4) << 3);
      rec[(size_t)row * NHv + ca] = acc0[e] + bra;
      rec[(size_t)row * NHv + cb] = acc1[e] + brb;
    }
    target += SCAN_WG; grid_barrier(bar, target);

    // ---- phase 2: gates + masked state update
    for (int idx = gtid; idx < Bv * Hv; idx += SCAN_WG * SCAN_THREADS) {
      const int b_ = idx >> 8;           // H == 256
      const int n  = idx & (Hv - 1);
      const bool mt = tokens[b_ * Tv + t] != 0;
      const float* xp = xproj + ((size_t)b_ * Tv + t) * NHv;
      const float* rp = rec + (size_t)b_ * NHv;
      const float xz = xp[n], xr = xp[Hv + n], xh = xp[2 * Hv + n];
      const float rz = rp[n], rr = rp[Hv + n], rh = rp[2 * Hv + n];
      const float hp = ldsH[b_ * HP + n];     // pre-update h (local LDS copy)
      const float z  = 1.0f / (1.0f + expf(-(xz + rz)));
      const float r  = 1.0f / (1.0f + expf(-(xr + rr)));
      const float hh = tanhf(xh + r * rh);
      const float hn = z * hp + (1.0f - z) * hh;
      const float hv = mt ? hn : hp;
      h[idx] = hv;
      xout[((size_t)b_ * Tv + t) * Hv + n] = hv;   // hs (post-mask carry)
    }
    target += SCAN_WG; grid_barrier(bar, target);
  }
}

// ------------------------------------------------------------------ FC head
__global__ void fc_kernel(const float* __restrict__ h,
                          const float* __restrict__ fcW,
                          const float* __restrict__ fcb,
                          float* __restrict__ out) {
  const int idx = blockIdx.x * blockDim.x + threadIdx.x;
  if (idx >= Bv * Cv) return;
  const int b = idx / Cv, c = idx % Cv;
  float s = fcb[c];
  for (int k = 0; k < Hv; ++k) s += h[b * Hv + k] * fcW[k * Cv + c];
  out[idx] = s;
}

// ------------------------------------------------------------------- launch
extern "C" void kernel_launch(void* const* d_in, const int* in_sizes, int n_in,
                              void* d_out, int out_size, void* d_ws,
                              size_t ws_size, hipStream_t stream) {
  const int*   tokens = (const int*)d_in[0];
  const float* emb    = (const float*)d_in[1];
  const float* W      = (const float*)d_in[2];   // [L, 256, 768]
  const float* Rm     = (const float*)d_in[3];   // [L, 256, 768]
  const float* b_i    = (const float*)d_in[4];   // [L, 768]
  const float* b_r    = (const float*)d_in[5];   // [L, 768]
  const float* fcW    = (const float*)d_in[6];   // [256, 10]
  const float* fcb    = (const float*)d_in[7];   // [10]
  float* out = (float*)d_out;

  float* ws  = (float*)d_ws;
  float* x0  = ws;                               // [B*T, 256]
  float* x1  = x0 + (size_t)Bv * Tv * Dv;        // [B*T, 256]
  float* xp  = x1 + (size_t)Bv * Tv * Hv;        // [B*T, 768]
  float* h   = xp + (size_t)Bv * Tv * NHv;       // [64, 256]
  float* rec = h + Bv * Hv;                      // [64, 768]
  unsigned* bar = (unsigned*)(rec + Bv * NHv);

  embed_kernel<<<Bv * Tv, 64, 0, stream>>>(tokens, emb, x0);

  const float* xin = x0;
  float* xout = x1;
  for (int l = 0; l < 5; ++l) {
    // (B*T/64) m-blocks * 48 n-tiles = 24576 waves / 8 per WG = 3072 WGs
    xproj_gemm_kernel<<<3072, 256, 0, stream>>>(
        xin, W + (size_t)l * Dv * NHv, b_i + (size_t)l * NHv, xp);
    bar_init_kernel<<<1, 1, 0, stream>>>(bar);
    gru_scan_kernel<<<SCAN_WG, SCAN_THREADS, 0, stream>>>(
        xp, Rm + (size_t)l * Hv * NHv, b_r + (size_t)l * NHv, tokens, h, rec,
        xout, bar);
    const float* t = xout; xout = (float*)xin; xin = t;   // ping-pong
  }
  fc_kernel<<<3, 256, 0, stream>>>(h, fcW, fcb, out);
}